// ConvolutionalCapsules_11072425689231
// MI455X (gfx1250) — compile-verified
//
#include <hip/hip_runtime.h>
#include <math.h>

typedef __attribute__((ext_vector_type(2))) float v2f;
typedef __attribute__((ext_vector_type(8))) float v8f;
typedef __attribute__((ext_vector_type(4))) int   v4i;

#define IMGS   32      // B*Nin
#define CIN    64      // D_IN * G
#define COUT   512     // NUM_OUT * D_OUT * G
#define WT_FLOATS (9 * 64 * 512)

// ---- CDNA5 async global->LDS staging (builtins exist per round-2 diagnostics:
//      b128: (addrspace(1) int4*, addrspace(3) int4*, imm, imm)
//      b32 : (addrspace(1) int*,  addrspace(3) int*,  imm, imm)
#if defined(__has_builtin)
#  if __has_builtin(__builtin_amdgcn_global_load_async_to_lds_b128) && \
      __has_builtin(__builtin_amdgcn_global_load_async_to_lds_b32)
#    define HAVE_ASYNC_LDS 1
#  endif
#  if __has_builtin(__builtin_amdgcn_s_wait_asynccnt)
#    define HAVE_WAIT_ASYNC 1
#  endif
#endif

typedef __attribute__((address_space(1))) v4i GI4;
typedef __attribute__((address_space(3))) v4i LI4;
typedef __attribute__((address_space(1))) int GI1;
typedef __attribute__((address_space(3))) int LI1;

static __device__ __forceinline__ GI4* gcast4(const void* p) { return (GI4*)(uintptr_t)p; }
static __device__ __forceinline__ LI4* lcast4(void* p)       { return (LI4*)(uint32_t)(uintptr_t)p; }
static __device__ __forceinline__ GI1* gcast1(const void* p) { return (GI1*)(uintptr_t)p; }
static __device__ __forceinline__ LI1* lcast1(void* p)       { return (LI1*)(uint32_t)(uintptr_t)p; }

#if defined(HAVE_ASYNC_LDS)
#  define ASYNC_B128(src, dst) \
     __builtin_amdgcn_global_load_async_to_lds_b128(gcast4(src), lcast4(dst), 0, 0)
#  define ASYNC_B32(src, dst) \
     __builtin_amdgcn_global_load_async_to_lds_b32(gcast1(src), lcast1(dst), 0, 0)
#else
#  define ASYNC_B128(src, dst) \
     do { *(float4*)(dst) = *(const float4*)(src); } while (0)
#  define ASYNC_B32(src, dst) \
     do { *(float*)(dst) = *(const float*)(src); } while (0)
#endif

#if defined(HAVE_WAIT_ASYNC)
#  define WAIT_ASYNC(n) __builtin_amdgcn_s_wait_asynccnt(n)
#else
#  define WAIT_ASYNC(n) asm volatile("s_wait_asynccnt " #n ::: "memory")
#endif

// ---------------------------------------------------------------------------
// Kernel 1: build the p4->p4 rotated/rolled weight bank.
// Output layout: wt[tap t (0..8)][ic (0..63)][oc (0..511)]  (oc fastest)
// wt[t][ci*4+s][co*4+r] = rot90^r( w[co][ci][(s-r)&3] )[kh][kw],  t = kh*3+kw
// ---------------------------------------------------------------------------
__global__ void prep_weights_kernel(const float* __restrict__ w,
                                    float* __restrict__ wt) {
  int idx = blockIdx.x * 256 + threadIdx.x;
  if (idx >= WT_FLOATS) return;
  int oc = idx & 511;
  int ic = (idx >> 9) & 63;
  int t  = idx >> 15;            // 0..8
  int kh = t / 3, kw = t - kh * 3;
  int r = oc & 3, co = oc >> 2;
  int s = ic & 3, ci = ic >> 2;
  int ssrc = (s - r) & 3;
  int si, sj;
  if (r == 0)      { si = kh;     sj = kw;     }
  else if (r == 1) { si = kw;     sj = 2 - kh; }   // rot90 ccw once
  else if (r == 2) { si = 2 - kh; sj = 2 - kw; }
  else             { si = 2 - kw; sj = kh;     }
  wt[idx] = w[((co * 16 + ci) * 4 + ssrc) * 9 + si * 3 + sj];
}

// ---------------------------------------------------------------------------
// Kernel 2: implicit-GEMM conv via V_WMMA_F32_16X16X4_F32, with async
// double-buffered LDS staging.  18 pipeline stages = 9 taps x 2 K-halves.
// Grid: x = image(32) * posblock(16) [64 positions], y = 4 channel blocks.
// Block = 256 threads = 8 waves; each wave: 16 ch x 64 pos -> 4 accumulators.
// ---------------------------------------------------------------------------
#define AS_STRIDE 136   // 2*136 % 64 == 16 -> half-waves hit disjoint banks
#define BS_STRIDE 72    // 2*72  % 64 == 16

__global__ __launch_bounds__(256) void conv_wmma_kernel(
    const float* __restrict__ x,    // (32, 64, 32, 32)
    const float* __restrict__ wt,   // (9, 64, 512)
    const float* __restrict__ bias, // (128,)
    float* __restrict__ u) {        // (32, 512, 32, 32)
  __shared__ __align__(16) float As[2][32 * AS_STRIDE];
  __shared__ __align__(16) float Bs[2][32 * BS_STRIDE];

  const int tid  = threadIdx.x;
  const int lane = tid & 31;
  const int wv   = tid >> 5;        // 0..7
  const int m    = lane & 15;
  const int half = lane >> 4;       // selects K pair (ISA A/B lane layout)
  const int bx   = blockIdx.x;
  const int n    = bx >> 4;         // image
  const int pb   = bx & 15;         // position block (2 rows of 32)
  const int h0   = pb * 2;
  const int oc_base = blockIdx.y * 128;

  v8f acc0 = {}, acc1 = {}, acc2 = {}, acc3 = {};

  // Stage chunk c (tap = c>>1, K-half = (c&1)*32) into buffer `buf`.
  // Exactly 12 async ops per wave per stage (4x b128 A + 8x b32 B): border
  // B elements load a clamped-but-valid address and are zeroed in fixup().
  auto stage = [&](int c, int buf) {
    const int tap = c >> 1;
    const int kh2 = (c & 1) << 5;
    const int dh  = tap / 3 - 1;
    const int dw  = tap - (tap / 3) * 3 - 1;
#pragma unroll
    for (int rr = 0; rr < 4; ++rr) {          // A: 32K x 128ch, b128 each
      int flat = tid * 4 + rr * 1024;
      int kl = flat >> 7, cch = flat & 127;
      const float* src = &wt[((tap * 64 + kh2 + kl) << 9) + oc_base + cch];
      ASYNC_B128(src, &As[buf][kl * AS_STRIDE + cch]);
    }
#pragma unroll
    for (int rr = 0; rr < 8; ++rr) {          // B: 32K x 64pos, b32 each
      int flat = tid + rr * 256;
      int kl = flat >> 6, p = flat & 63;
      int hh = h0 + (p >> 5) + dh;
      int ww = (p & 31) + dw;
      int hhc = min(max(hh, 0), 31);
      int wwc = min(max(ww, 0), 31);
      const float* src = &x[(((n << 6) + kh2 + kl) << 10) + (hhc << 5) + wwc];
      ASYNC_B32(src, &Bs[buf][kl * BS_STRIDE + p]);
    }
  };
  auto fixup = [&](int c, int buf) {          // zero the padded border elems
    const int tap = c >> 1;
    const int dh  = tap / 3 - 1;
    const int dw  = tap - (tap / 3) * 3 - 1;
#pragma unroll
    for (int rr = 0; rr < 8; ++rr) {
      int flat = tid + rr * 256;
      int kl = flat >> 6, p = flat & 63;
      int hh = h0 + (p >> 5) + dh;
      int ww = (p & 31) + dw;
      if (!((unsigned)hh < 32u && (unsigned)ww < 32u))
        Bs[buf][kl * BS_STRIDE + p] = 0.f;
    }
  };

  stage(0, 0);
  for (int c = 0; c < 18; ++c) {
    const int buf = c & 1;
    if (c < 17) {
      stage(c + 1, buf ^ 1);
      WAIT_ASYNC(12);        // stage c complete (stage c+1 contributes 12)
    } else {
      WAIT_ASYNC(0);
    }
    fixup(c, buf);
    __syncthreads();

    const float* as = As[buf];
    const float* bs = Bs[buf];
#pragma unroll
    for (int k0 = 0; k0 < 32; k0 += 4) {
      const int kr = k0 + half * 2;   // lanes 16-31 hold K+2,K+3 (ISA layout)
      v2f a;
      a.x = as[kr * AS_STRIDE + wv * 16 + m];
      a.y = as[(kr + 1) * AS_STRIDE + wv * 16 + m];
      v2f b0, b1, b2, b3;
      b0.x = bs[kr * BS_STRIDE +  0 + m]; b0.y = bs[(kr + 1) * BS_STRIDE +  0 + m];
      b1.x = bs[kr * BS_STRIDE + 16 + m]; b1.y = bs[(kr + 1) * BS_STRIDE + 16 + m];
      b2.x = bs[kr * BS_STRIDE + 32 + m]; b2.y = bs[(kr + 1) * BS_STRIDE + 32 + m];
      b3.x = bs[kr * BS_STRIDE + 48 + m]; b3.y = bs[(kr + 1) * BS_STRIDE + 48 + m];
      acc0 = __builtin_amdgcn_wmma_f32_16x16x4_f32(false, a, false, b0, (short)0, acc0, false, false);
      acc1 = __builtin_amdgcn_wmma_f32_16x16x4_f32(false, a, false, b1, (short)0, acc1, false, false);
      acc2 = __builtin_amdgcn_wmma_f32_16x16x4_f32(false, a, false, b2, (short)0, acc2, false, false);
      acc3 = __builtin_amdgcn_wmma_f32_16x16x4_f32(false, a, false, b3, (short)0, acc3, false, false);
    }
    __syncthreads();   // all waves done with buf before it is restaged
  }

  // D layout: VGPR v -> rows M=v (lanes 0-15) / M=v+8 (lanes 16-31), N = lane&15
#pragma unroll
  for (int v = 0; v < 8; ++v) {
    int oc = oc_base + wv * 16 + v + half * 8;
    float bb = bias[oc >> 2];
    size_t rowbase = ((size_t)(n * COUT + oc) << 10) + (size_t)(pb * 64 + m);
    u[rowbase +  0] = acc0[v] + bb;
    u[rowbase + 16] = acc1[v] + bb;
    u[rowbase + 32] = acc2[v] + bb;
    u[rowbase + 48] = acc3[v] + bb;
  }
}

// ---------------------------------------------------------------------------
// Kernel 3: LayerNorm + cosine-routing + squash. One thread per
// (b, nout, h, w) site; loops over g and nin fully unrolled (register tiles).
// ---------------------------------------------------------------------------
__global__ __launch_bounds__(256) void routing_kernel(
    const float* __restrict__ u,      // (32, 512, 32, 32)
    const float* __restrict__ gamma,  // (16,)
    const float* __restrict__ beta,   // (16,)
    float* __restrict__ out) {        // (4, 8, 16, 4, 32, 32)
  int idx  = blockIdx.x * 256 + threadIdx.x;   // 32768 sites
  int pos  = idx & 1023;
  int nout = (idx >> 10) & 7;
  int b    = idx >> 13;

  float gm[16], bt[16];
#pragma unroll
  for (int d = 0; d < 16; ++d) { gm[d] = gamma[d]; bt[d] = beta[d]; }

  float s[4][16];
#pragma unroll
  for (int g = 0; g < 4; ++g) {
    float up[8][16];
#pragma unroll
    for (int i = 0; i < 8; ++i) {
      // channel(nout,d,g) = nout*64 + d*4 + g within image b*8+i
      const float* base =
          u + (((size_t)(b * 8 + i) * 512 + (nout << 6) + g) << 10) + pos;
      float mu = 0.f;
#pragma unroll
      for (int d = 0; d < 16; ++d) { up[i][d] = base[(size_t)d * 4096]; mu += up[i][d]; }
      mu *= 0.0625f;
      float var = 0.f;
#pragma unroll
      for (int d = 0; d < 16; ++d) { float z = up[i][d] - mu; var += z * z; }
      var *= 0.0625f;
      float inv = 1.0f / sqrtf(var + 1e-5f);
#pragma unroll
      for (int d = 0; d < 16; ++d) up[i][d] = (up[i][d] - mu) * inv * gm[d] + bt[d];
    }
    // sum_j dot(up_i, up_j) == dot(up_i, colsum)
    float cs[16];
#pragma unroll
    for (int d = 0; d < 16; ++d) {
      float a = 0.f;
#pragma unroll
      for (int i = 0; i < 8; ++i) a += up[i][d];
      cs[d] = a;
    }
    float rowsum[8];
#pragma unroll
    for (int i = 0; i < 8; ++i) {
      float nsq = 0.f, rs = 0.f;
#pragma unroll
      for (int d = 0; d < 16; ++d) { nsq += up[i][d] * up[i][d]; rs += up[i][d] * cs[d]; }
      rowsum[i] = rs / fmaxf(nsq, 1e-8f);
    }
    float mx = rowsum[0];
#pragma unroll
    for (int i = 1; i < 8; ++i) mx = fmaxf(mx, rowsum[i]);
    float e[8], Z = 0.f;
#pragma unroll
    for (int i = 0; i < 8; ++i) { e[i] = expf(rowsum[i] - mx); Z += e[i]; }
    float invZ = 1.f / Z;
#pragma unroll
    for (int d = 0; d < 16; ++d) {
      float a = 0.f;
#pragma unroll
      for (int i = 0; i < 8; ++i) a += e[i] * up[i][d];
      s[g][d] = a * invZ;
    }
  }
  // squash over g; out[(((b*8+nout)*16 + d)*4 + g)*1024 + pos]
  size_t obase = ((size_t)(b * 8 + nout) << 16) + pos;
#pragma unroll
  for (int d = 0; d < 16; ++d) {
    float nsq = s[0][d] * s[0][d] + s[1][d] * s[1][d] +
                s[2][d] * s[2][d] + s[3][d] * s[3][d];
    float scale = (nsq / (1.f + nsq)) / sqrtf(nsq + 1e-16f);
#pragma unroll
    for (int g = 0; g < 4; ++g)
      out[obase + ((size_t)(d * 4 + g) << 10)] = s[g][d] * scale;
  }
}

// ---------------------------------------------------------------------------
extern "C" void kernel_launch(void* const* d_in, const int* in_sizes, int n_in,
                              void* d_out, int out_size, void* d_ws, size_t ws_size,
                              hipStream_t stream) {
  const float* caps  = (const float*)d_in[0];  // (4,8,16,4,32,32)
  const float* convw = (const float*)d_in[1];  // (128,16,4,3,3)
  const float* convb = (const float*)d_in[2];  // (128,)
  const float* gamma = (const float*)d_in[3];  // (16,)
  const float* beta  = (const float*)d_in[4];  // (16,)

  float* wt = (float*)d_ws;                    // 9*64*512 floats (1.125 MiB)
  float* u  = wt + WT_FLOATS;                  // 32*512*1024 floats (64 MiB)
  float* out = (float*)d_out;

  prep_weights_kernel<<<(WT_FLOATS + 255) / 256, 256, 0, stream>>>(convw, wt);
  conv_wmma_kernel<<<dim3(32 * 16, 4), 256, 0, stream>>>(caps, wt, convb, u);
  routing_kernel<<<(4 * 8 * 1024) / 256, 256, 0, stream>>>(u, gamma, beta, out);
}